// ToRGBLayer_65558380806669
// MI455X (gfx1250) — compile-verified
//
#include <hip/hip_runtime.h>

// ---- problem constants (match reference) ----
#define W_DIM 512
#define IN_CH 512
#define OUT_CH 3
#define HW    16384      // 128*128
#define BATCH 32
#define CLAMP_V 256.0f

typedef __attribute__((ext_vector_type(2))) float v2f;
typedef __attribute__((ext_vector_type(4))) float v4f;
typedef __attribute__((ext_vector_type(8))) float v8f;

// -----------------------------------------------------------------------------
// Kernel 1: styles = (w @ (affine_weight * fc_gain)^T + affine_bias) * weight_gain
// 32x512 = (32x512) x (512x512)^T, f32 GEMM via V_WMMA_F32_16X16X4_F32.
// One wave (32 threads) per 16x16 output tile; grid = (512/16, 32/16) = (32, 2).
// A layout (16x4 f32): lane L holds A[L%16][k0 + 2*(L/16) + {0,1}] in 2 VGPRs.
// B layout (4x16 f32): lane L holds B[k0 + 2*(L/16) + {0,1}][L%16]; since
//   B = aw^T, B[k][n] = aw[n][k] -> same row-major addressing as A but on aw.
// C/D layout: vgpr r, lane L -> element (M = r + 8*(L/16), N = L%16).
// -----------------------------------------------------------------------------
__global__ __launch_bounds__(32)
void styles_wmma_kernel(const float* __restrict__ w,    // (32, 512)
                        const float* __restrict__ aw,   // (512, 512)
                        const float* __restrict__ ab,   // (512)
                        float* __restrict__ styles)     // (32, 512)
{
    const int lane  = threadIdx.x & 31;
    const int l16   = lane & 15;
    const int half  = lane >> 4;           // 0 or 1
    const int nBase = blockIdx.x * 16;     // style / in-channel columns
    const int mBase = blockIdx.y * 16;     // batch rows

    const float* __restrict__ aRow = w  + (size_t)(mBase + l16) * W_DIM + 2 * half;
    const float* __restrict__ bRow = aw + (size_t)(nBase + l16) * W_DIM + 2 * half;

    v8f c = {0.f, 0.f, 0.f, 0.f, 0.f, 0.f, 0.f, 0.f};
    for (int k0 = 0; k0 < W_DIM; k0 += 4) {
        v2f a, b;
        a.x = aRow[k0];
        a.y = aRow[k0 + 1];
        b.x = bRow[k0];
        b.y = bRow[k0 + 1];
        // (neg_a, A, neg_b, B, c_mod, C, reuse_a, reuse_b)
        c = __builtin_amdgcn_wmma_f32_16x16x4_f32(false, a, false, b,
                                                  (short)0, c, false, false);
    }

    const float fc_gain = 0.04419417382415922f;   // 1/sqrt(512)
    const float w_gain  = 0.04419417382415922f;   // 1/sqrt(512)
    const float abv = ab[nBase + l16];

#pragma unroll
    for (int r = 0; r < 8; ++r) {
        const int m = mBase + r + 8 * half;
        styles[(size_t)m * IN_CH + nBase + l16] = (c[r] * fc_gain + abv) * w_gain;
    }
}

// -----------------------------------------------------------------------------
// Kernel 2: out[b,o,p] = clamp( sum_i x[b,i,p] * (wk[o,i]*styles[b,i]) + bias[o] )
// Pure HBM stream of x (1.07 GB read once) -> bandwidth bound (~46 us floor at
// 23.3 TB/s). Each thread owns one float4 of pixels, loops 512 channels with
// b128 nontemporal loads; modulated weights broadcast from 8 KB LDS.
// grid = (HW / 1024, 32), block = 256 threads (8 waves).
// -----------------------------------------------------------------------------
#define TPB 256

__device__ __forceinline__ v4f clamp_bias(v4f v, float bo) {
    v4f r;
    r.x = fminf(fmaxf(v.x + bo, -CLAMP_V), CLAMP_V);
    r.y = fminf(fmaxf(v.y + bo, -CLAMP_V), CLAMP_V);
    r.z = fminf(fmaxf(v.z + bo, -CLAMP_V), CLAMP_V);
    r.w = fminf(fmaxf(v.w + bo, -CLAMP_V), CLAMP_V);
    return r;
}

__global__ __launch_bounds__(TPB)
void modrgb_kernel(const float* __restrict__ x,       // (32, 512, 128, 128)
                   const float* __restrict__ styles,  // (32, 512)
                   const float* __restrict__ wk,      // (3, 512) [= weight[:,:,0,0]]
                   const float* __restrict__ bias,    // (3)
                   float* __restrict__ out)           // (32, 3, 128, 128)
{
    __shared__ v4f sm[IN_CH];   // per channel: (m0, m1, m2, 0)

    const int b   = blockIdx.y;
    const int tid = threadIdx.x;

    // Build modulated weights for this batch element into LDS.
    for (int i = tid; i < IN_CH; i += TPB) {
        const float st = styles[(size_t)b * IN_CH + i];
        v4f v;
        v.x = wk[i]             * st;
        v.y = wk[IN_CH + i]     * st;
        v.z = wk[2 * IN_CH + i] * st;
        v.w = 0.f;
        sm[i] = v;
    }
    __syncthreads();

    const int pix = blockIdx.x * (TPB * 4) + tid * 4;
    const v4f* __restrict__ xp =
        (const v4f*)(x + (size_t)b * IN_CH * HW + pix);

    v4f a0 = {0.f, 0.f, 0.f, 0.f};
    v4f a1 = {0.f, 0.f, 0.f, 0.f};
    v4f a2 = {0.f, 0.f, 0.f, 0.f};

#pragma unroll 8
    for (int i = 0; i < IN_CH; ++i) {
        const v4f xv = __builtin_nontemporal_load(&xp[(size_t)i * (HW / 4)]);
        const v4f s  = sm[i];   // uniform address -> LDS broadcast
        a0.x = fmaf(xv.x, s.x, a0.x); a0.y = fmaf(xv.y, s.x, a0.y);
        a0.z = fmaf(xv.z, s.x, a0.z); a0.w = fmaf(xv.w, s.x, a0.w);
        a1.x = fmaf(xv.x, s.y, a1.x); a1.y = fmaf(xv.y, s.y, a1.y);
        a1.z = fmaf(xv.z, s.y, a1.z); a1.w = fmaf(xv.w, s.y, a1.w);
        a2.x = fmaf(xv.x, s.z, a2.x); a2.y = fmaf(xv.y, s.z, a2.y);
        a2.z = fmaf(xv.z, s.z, a2.z); a2.w = fmaf(xv.w, s.z, a2.w);
    }

    const float b0 = bias[0], b1 = bias[1], b2 = bias[2];
    float* obase = out + (size_t)b * OUT_CH * HW + pix;
    __builtin_nontemporal_store(clamp_bias(a0, b0), (v4f*)(obase));
    __builtin_nontemporal_store(clamp_bias(a1, b1), (v4f*)(obase + HW));
    __builtin_nontemporal_store(clamp_bias(a2, b2), (v4f*)(obase + 2 * HW));
}

// -----------------------------------------------------------------------------
// Launch: inputs in setup_inputs() order:
//   d_in[0]=x, d_in[1]=w, d_in[2]=affine_weight, d_in[3]=affine_bias,
//   d_in[4]=weight(3,512,1,1), d_in[5]=bias(3). All float32. Output float32.
// d_ws holds styles (32*512 floats = 64 KB).
// -----------------------------------------------------------------------------
extern "C" void kernel_launch(void* const* d_in, const int* in_sizes, int n_in,
                              void* d_out, int out_size, void* d_ws, size_t ws_size,
                              hipStream_t stream) {
    const float* x    = (const float*)d_in[0];
    const float* w    = (const float*)d_in[1];
    const float* aw   = (const float*)d_in[2];
    const float* ab   = (const float*)d_in[3];
    const float* wk   = (const float*)d_in[4];
    const float* bias = (const float*)d_in[5];
    float* out    = (float*)d_out;
    float* styles = (float*)d_ws;   // 32*512 f32

    // Tiny WMMA GEMM for styles.
    styles_wmma_kernel<<<dim3(W_DIM / 16, BATCH / 16), 32, 0, stream>>>(
        w, aw, ab, styles);

    // Bandwidth-bound modulated 1x1 conv + bias + clamp.
    modrgb_kernel<<<dim3(HW / (TPB * 4), BATCH), TPB, 0, stream>>>(
        x, styles, wk, bias, out);
}